// BlockTopkDecoderLayer_17832704213649
// MI455X (gfx1250) — compile-verified
//
#include <hip/hip_runtime.h>
#include <hip/hip_bf16.h>

// ---------------------------------------------------------------------------
// BlockTopkDecoderLayer on MI455X (gfx1250), f32 end-to-end.
// B=256, D=4096, H=4, DH=128, KVH=1, I=1536, PAGE=16, NPAGES=64, TOPK=16
// GEMMs use V_WMMA_F32_16X16X4_F32 (wave32). Memory-bound: weights streamed
// once per launch; W tiles staged in LDS and shared by 8 waves per block.
// ---------------------------------------------------------------------------

#define B_   256
#define D_   4096
#define H_   4
#define DH_  128
#define I_   1536
#define PAGE_ 16
#define NPAGES_ 64
#define TOPK_ 16
#define EPS_ 1e-6f

typedef __attribute__((ext_vector_type(2))) float v2f;
typedef __attribute__((ext_vector_type(8))) float v8f;

// ---------------- utility reductions (wave32) ------------------------------
__device__ __forceinline__ float wave_sum(float v) {
#pragma unroll
  for (int o = 16; o > 0; o >>= 1) v += __shfl_xor(v, o, 32);
  return v;
}
__device__ __forceinline__ float wave_max(float v) {
#pragma unroll
  for (int o = 16; o > 0; o >>= 1) v = fmaxf(v, __shfl_xor(v, o, 32));
  return v;
}

// block reduce-sum for blockDim.x = 32*nwaves, nwaves <= 8
template <int NW>
__device__ __forceinline__ float block_sum(float v, float* scratch) {
  int lane = threadIdx.x & 31, w = threadIdx.x >> 5;
  v = wave_sum(v);
  if (lane == 0) scratch[w] = v;
  __syncthreads();
  float t = (threadIdx.x < NW) ? scratch[threadIdx.x] : 0.0f;
  if (w == 0) {
#pragma unroll
    for (int o = NW >> 1; o > 0; o >>= 1) t += __shfl_xor(t, o, 32);
    if (lane == 0) scratch[0] = t;
  }
  __syncthreads();
  float r = scratch[0];
  __syncthreads();
  return r;
}

// ---------------- RMSNorm over D=4096 (one block per row) ------------------
__global__ __launch_bounds__(256) void rmsnorm_kernel(
    const float* __restrict__ x, const float* __restrict__ w,
    float* __restrict__ out) {
  __shared__ float scr[8];
  int row = blockIdx.x;
  const float* xr = x + (size_t)row * D_;
  float ss = 0.0f;
  for (int i = threadIdx.x; i < D_; i += 256) {
    float v = xr[i];
    ss += v * v;
  }
  ss = block_sum<8>(ss, scr);
  float r = rsqrtf(ss * (1.0f / D_) + EPS_);
  float* orow = out + (size_t)row * D_;
  for (int i = threadIdx.x; i < D_; i += 256) orow[i] = xr[i] * r * w[i];
}

// ---------------- WMMA f32 GEMM: C[M,N] = A[M,K] @ W[K,N] (+bias)(+res) ----
// grid = (N/16, M/128), block = 256 (8 waves). Each wave: one 16x16 tile,
// m0 = by*128 + wave*16, n0 = bx*16. W chunk (64xK-slab x 16) staged in LDS,
// shared by all 8 waves. A read from global (L2-resident).
__global__ __launch_bounds__(256) void gemm16_wmma(
    const float* __restrict__ A, const float* __restrict__ W,
    const float* __restrict__ bias, const float* __restrict__ res,
    float* __restrict__ C, int M, int N, int K) {
  __shared__ __align__(16) float lwB[64][16];
  const int n0 = blockIdx.x * 16;
  const int wave = threadIdx.x >> 5;
  const int lane = threadIdx.x & 31;
  const int lane16 = lane & 15;
  const int half = lane >> 4;      // 0: K{0,1} / rows r, 1: K{2,3} / rows r+8
  const int koff = half * 2;
  const int m0 = blockIdx.y * 128 + wave * 16;

  const int ldrow = threadIdx.x >> 2;        // 0..63
  const int ldcol = (threadIdx.x & 3) * 4;   // 0,4,8,12

  v8f acc = {0.f, 0.f, 0.f, 0.f, 0.f, 0.f, 0.f, 0.f};
  const float* Arow = A + (size_t)(m0 + lane16) * K;

  for (int k0 = 0; k0 < K; k0 += 64) {
    __syncthreads();
    // cooperative stage of W[k0..k0+63][n0..n0+15] (float4 per thread)
    float4 wv = *(const float4*)(W + (size_t)(k0 + ldrow) * N + n0 + ldcol);
    *(float4*)&lwB[ldrow][ldcol] = wv;
    // prefetch next W slab into caches (global_prefetch_b8)
    if (k0 + 64 < K)
      __builtin_prefetch(W + (size_t)(k0 + 64 + ldrow) * N + n0 + ldcol, 0, 1);
    __syncthreads();
#pragma unroll
    for (int kk = 0; kk < 64; kk += 4) {
      v2f a = *(const v2f*)(Arow + k0 + kk + koff);             // A[M][K..K+1]
      v2f b = {lwB[kk + koff][lane16], lwB[kk + koff + 1][lane16]};
      acc = __builtin_amdgcn_wmma_f32_16x16x4_f32(
          false, a, false, b, (short)0, acc, false, false);
    }
  }

  float bv = bias ? bias[n0 + lane16] : 0.0f;
#pragma unroll
  for (int r = 0; r < 8; ++r) {
    int row = m0 + r + half * 8;
    size_t idx = (size_t)row * N + n0 + lane16;
    float v = acc[r] + bv;
    if (res) v += res[idx];
    C[idx] = v;
  }
}

// ---------------- fused gate/up GEMM + SiLU*up -----------------------------
// act[M,I] = silu(A@gate_w) * (A@up_w).  grid=(I/16, M/128), block=256.
__global__ __launch_bounds__(256) void gemm_gateup_wmma(
    const float* __restrict__ A, const float* __restrict__ Wg,
    const float* __restrict__ Wu, float* __restrict__ act, int K, int N) {
  __shared__ __align__(16) float lwG[64][16];
  __shared__ __align__(16) float lwU[64][16];
  const int n0 = blockIdx.x * 16;
  const int wave = threadIdx.x >> 5;
  const int lane = threadIdx.x & 31;
  const int lane16 = lane & 15;
  const int half = lane >> 4;
  const int koff = half * 2;
  const int m0 = blockIdx.y * 128 + wave * 16;
  const int ldrow = threadIdx.x >> 2;
  const int ldcol = (threadIdx.x & 3) * 4;

  v8f accG = {0.f, 0.f, 0.f, 0.f, 0.f, 0.f, 0.f, 0.f};
  v8f accU = {0.f, 0.f, 0.f, 0.f, 0.f, 0.f, 0.f, 0.f};
  const float* Arow = A + (size_t)(m0 + lane16) * K;

  for (int k0 = 0; k0 < K; k0 += 64) {
    __syncthreads();
    *(float4*)&lwG[ldrow][ldcol] =
        *(const float4*)(Wg + (size_t)(k0 + ldrow) * N + n0 + ldcol);
    *(float4*)&lwU[ldrow][ldcol] =
        *(const float4*)(Wu + (size_t)(k0 + ldrow) * N + n0 + ldcol);
    if (k0 + 64 < K) {
      __builtin_prefetch(Wg + (size_t)(k0 + 64 + ldrow) * N + n0 + ldcol, 0, 1);
      __builtin_prefetch(Wu + (size_t)(k0 + 64 + ldrow) * N + n0 + ldcol, 0, 1);
    }
    __syncthreads();
#pragma unroll
    for (int kk = 0; kk < 64; kk += 4) {
      v2f a = *(const v2f*)(Arow + k0 + kk + koff);
      v2f bg = {lwG[kk + koff][lane16], lwG[kk + koff + 1][lane16]};
      v2f bu = {lwU[kk + koff][lane16], lwU[kk + koff + 1][lane16]};
      accG = __builtin_amdgcn_wmma_f32_16x16x4_f32(
          false, a, false, bg, (short)0, accG, false, false);
      accU = __builtin_amdgcn_wmma_f32_16x16x4_f32(
          false, a, false, bu, (short)0, accU, false, false);
    }
  }
#pragma unroll
  for (int r = 0; r < 8; ++r) {
    int row = m0 + r + half * 8;
    size_t idx = (size_t)row * N + n0 + lane16;
    float g = accG[r], u = accU[r];
    act[idx] = (g / (1.0f + __expf(-g))) * u;  // silu(g) * u
  }
}

// ---------------- q/k RMSNorm (per-head DH=128), block per batch row -------
__global__ __launch_bounds__(128) void qknorm_kernel(
    const float* __restrict__ proj, const float* __restrict__ qnw,
    const float* __restrict__ knw, float* __restrict__ qn,
    float* __restrict__ kn) {
  __shared__ float scr[4];
  int b = blockIdx.x, t = threadIdx.x;  // t = d in [0,128)
  // k
  {
    float kv = proj[(size_t)b * 768 + 512 + t];
    float ss = block_sum<4>(kv * kv, scr);
    float r = rsqrtf(ss * (1.0f / DH_) + EPS_);
    kn[(size_t)b * DH_ + t] = kv * r * knw[t];
  }
  // q heads
  for (int h = 0; h < H_; ++h) {
    float qv = proj[(size_t)b * 768 + h * DH_ + t];
    float ss = block_sum<4>(qv * qv, scr);
    float r = rsqrtf(ss * (1.0f / DH_) + EPS_);
    qn[(size_t)b * (H_ * DH_) + h * DH_ + t] = qv * r * qnw[t];
  }
}

// ---------------- compressed-key similarity + top-16 of 64 -----------------
__global__ __launch_bounds__(128) void topk_kernel(
    const float* __restrict__ qn, const float* __restrict__ cck,
    int* __restrict__ topk) {
  __shared__ float s_qsum[DH_];
  __shared__ float s_sim[NPAGES_];
  int b = blockIdx.x, t = threadIdx.x;
  float qs = 0.0f;
#pragma unroll
  for (int h = 0; h < H_; ++h) qs += qn[(size_t)b * 512 + h * DH_ + t];
  s_qsum[t] = qs;
  __syncthreads();
  if (t < NPAGES_) {
    const float* row = cck + ((size_t)b * NPAGES_ + t) * DH_;
    float dot = 0.0f;
    for (int d = 0; d < DH_; ++d) dot += row[d] * s_qsum[d];
    s_sim[t] = dot * 0.25f;  // mean over H (scaling irrelevant to ranking)
  }
  __syncthreads();
  if (t == 0) {
    unsigned long long used = 0ull;
    for (int k = 0; k < TOPK_; ++k) {
      float best = -3.4e38f;
      int bi = 0;
      for (int n = 0; n < NPAGES_; ++n)
        if (!((used >> n) & 1ull) && s_sim[n] > best) { best = s_sim[n]; bi = n; }
      used |= 1ull << bi;
      topk[b * TOPK_ + k] = bi;
    }
  }
}

// ---------------- paged block-topk attention -------------------------------
// One block (256 thr, 8 waves) per batch row; all 4 heads.
// 17 pages x 16 slots = 272 kv slots; last page valid only at slot 0
// (freshly-written k,v for the current token).
__global__ __launch_bounds__(256) void attn_kernel(
    const float* __restrict__ qn, const float* __restrict__ kn,
    const float* __restrict__ proj, const float* __restrict__ kv,
    const int* __restrict__ topk, float* __restrict__ o) {
  __shared__ float s_q[H_][DH_];
  __shared__ float s_sc[H_][288];
  __shared__ int s_phys[TOPK_ + 1];
  const int b = blockIdx.x;
  const int wave = threadIdx.x >> 5, lane = threadIdx.x & 31;
  const float scale = 0.08838834764831845f;  // 1/sqrt(128)

  for (int i = threadIdx.x; i < H_ * DH_; i += 256)
    ((float*)s_q)[i] = qn[(size_t)b * (H_ * DH_) + i];
  if (threadIdx.x < TOPK_) s_phys[threadIdx.x] = topk[b * TOPK_ + threadIdx.x] * B_ + b;
  if (threadIdx.x == TOPK_) s_phys[TOPK_] = NPAGES_ * B_ + b;
  __syncthreads();

  // scores: wave-per-slot, lanes split DH (4 f32 each, coalesced float4)
  for (int s = wave; s < 272; s += 8) {
    int p = s >> 4, slot = s & 15;
    bool valid = (p < TOPK_) || (slot == 0);
    float a[H_] = {0.f, 0.f, 0.f, 0.f};
    if (valid) {
      const float* kptr =
          (p < TOPK_) ? kv + (size_t)s_phys[p] * 4096 + (size_t)slot * DH_
                      : kn + (size_t)b * DH_;
      float4 k4 = *(const float4*)(kptr + lane * 4);
      float kk[4] = {k4.x, k4.y, k4.z, k4.w};
#pragma unroll
      for (int h = 0; h < H_; ++h)
#pragma unroll
        for (int j = 0; j < 4; ++j) a[h] += kk[j] * s_q[h][lane * 4 + j];
    }
#pragma unroll
    for (int h = 0; h < H_; ++h) a[h] = wave_sum(a[h]);
    if (lane == 0) {
#pragma unroll
      for (int h = 0; h < H_; ++h)
        s_sc[h][s] = valid ? a[h] * scale : -1e30f;
    }
  }
  __syncthreads();

  // softmax, one wave per head
  if (wave < H_) {
    int h = wave;
    float m = -3.4e38f;
    for (int n = lane; n < 272; n += 32) m = fmaxf(m, s_sc[h][n]);
    m = wave_max(m);
    float sum = 0.0f;
    for (int n = lane; n < 272; n += 32) {
      float e = __expf(s_sc[h][n] - m);
      s_sc[h][n] = e;
      sum += e;
    }
    sum = wave_sum(sum);
    float inv = 1.0f / sum;
    for (int n = lane; n < 272; n += 32) s_sc[h][n] *= inv;
  }
  __syncthreads();

  // o[h][d] = sum_s prob[h][s] * val[s][d]; 64 threads per head, 2 d each
  {
    int h = threadIdx.x >> 6;
    int d0 = (threadIdx.x & 63) * 2;
    float acc0 = 0.f, acc1 = 0.f;
    for (int s = 0; s < 272; ++s) {
      int p = s >> 4, slot = s & 15;
      if (p >= TOPK_ && slot != 0) continue;  // prob == 0 exactly
      float pr = s_sc[h][s];
      const float* vptr =
          (p < TOPK_)
              ? kv + (size_t)s_phys[p] * 4096 + 2048 + (size_t)slot * DH_
              : proj + (size_t)b * 768 + 640;  // fresh v
      acc0 += pr * vptr[d0];
      acc1 += pr * vptr[d0 + 1];
    }
    o[(size_t)b * (H_ * DH_) + h * DH_ + d0] = acc0;
    o[(size_t)b * (H_ * DH_) + h * DH_ + d0 + 1] = acc1;
  }
}

// ---------------------------------------------------------------------------
extern "C" void kernel_launch(void* const* d_in, const int* in_sizes, int n_in,
                              void* d_out, int out_size, void* d_ws,
                              size_t ws_size, hipStream_t stream) {
  const float* hidden   = (const float*)d_in[0];
  const float* wqkv_w   = (const float*)d_in[1];
  const float* wqkv_b   = (const float*)d_in[2];
  const float* wo_w     = (const float*)d_in[3];
  const float* up_w     = (const float*)d_in[4];
  const float* gate_w   = (const float*)d_in[5];
  const float* down_w   = (const float*)d_in[6];
  const float* qnorm_w  = (const float*)d_in[7];
  const float* knorm_w  = (const float*)d_in[8];
  const float* in_ln_w  = (const float*)d_in[9];
  const float* post_ln_w= (const float*)d_in[10];
  const float* kv_cache = (const float*)d_in[11];
  const float* cck      = (const float*)d_in[12];

  float* ws = (float*)d_ws;
  float* w_h      = ws;                          // 256*4096
  float* w_proj   = w_h      + (size_t)B_ * D_;  // 256*768
  float* w_qn     = w_proj   + (size_t)B_ * 768; // 256*512
  float* w_kn     = w_qn     + (size_t)B_ * 512; // 256*128
  float* w_o      = w_kn     + (size_t)B_ * 128; // 256*512
  float* w_res2   = w_o      + (size_t)B_ * 512; // 256*4096
  float* w_h2     = w_res2   + (size_t)B_ * D_;  // 256*4096
  float* w_act    = w_h2     + (size_t)B_ * D_;  // 256*1536
  int*   w_topk   = (int*)(w_act + (size_t)B_ * I_);  // 256*16 ints

  // 1) h = rms(hidden, in_ln)
  rmsnorm_kernel<<<B_, 256, 0, stream>>>(hidden, in_ln_w, w_h);
  // 2) proj = h @ wqkv + bias   (M=256, N=768, K=4096)
  gemm16_wmma<<<dim3(768 / 16, 2), 256, 0, stream>>>(
      w_h, wqkv_w, wqkv_b, nullptr, w_proj, B_, 768, D_);
  // 3) per-head q/k RMSNorm
  qknorm_kernel<<<B_, 128, 0, stream>>>(w_proj, qnorm_w, knorm_w, w_qn, w_kn);
  // 4) compressed-key sim + top-16 pages
  topk_kernel<<<B_, 128, 0, stream>>>(w_qn, cck, w_topk);
  // 5) paged attention over 16 topk pages + fresh (k,v)
  attn_kernel<<<B_, 256, 0, stream>>>(w_qn, w_kn, w_proj, kv_cache, w_topk, w_o);
  // 6) resid2 = hidden + o @ wo   (M=256, N=4096, K=512)
  gemm16_wmma<<<dim3(D_ / 16, 2), 256, 0, stream>>>(
      w_o, wo_w, nullptr, hidden, w_res2, B_, D_, 512);
  // 7) h2 = rms(resid2, post_ln)
  rmsnorm_kernel<<<B_, 256, 0, stream>>>(w_res2, post_ln_w, w_h2);
  // 8) act = silu(h2@gate) * (h2@up)   (M=256, N=1536, K=4096)
  gemm_gateup_wmma<<<dim3(I_ / 16, 2), 256, 0, stream>>>(
      w_h2, gate_w, up_w, w_act, D_, I_);
  // 9) out = resid2 + act @ down   (M=256, N=4096, K=1536)
  gemm16_wmma<<<dim3(D_ / 16, 2), 256, 0, stream>>>(
      w_act, down_w, nullptr, w_res2, (float*)d_out, B_, D_, I_);
}